// DrugGNN_8804682957056
// MI455X (gfx1250) — compile-verified
//
#include <hip/hip_runtime.h>

typedef float v2f __attribute__((ext_vector_type(2)));
typedef float v8f __attribute__((ext_vector_type(8)));

#define N_NODES 200000
#define N_EDGES 3200000
#define N_GRAPH 2048
#define HID     64
#define BPAIR_STRIDE 80   // v2f units per K-pair row (=160 floats; 640B => half-waves hit disjoint banks)

// ---------------- degree / dinv ----------------
__global__ void k_init_deg(float* deg, int n) {
    int i = blockIdx.x * blockDim.x + threadIdx.x;
    if (i < n) deg[i] = 1.0f;                 // self-loop
}

__global__ void k_deg_edges(const int* __restrict__ dst, float* deg, int e) {
    int i = blockIdx.x * blockDim.x + threadIdx.x;
    if (i < e) atomicAdd(&deg[dst[i]], 1.0f);
}

// deg -> dinv in place; also seed layer-1 scalar sum with self-loop term
__global__ void k_dinv_selfloop(float* deg_dinv, const float* __restrict__ x,
                                float* s, int n) {
    int i = blockIdx.x * blockDim.x + threadIdx.x;
    if (i < n) {
        float d = rsqrtf(deg_dinv[i]);
        deg_dinv[i] = d;
        s[i] = d * d * x[i];
    }
}

// ---------------- layer 1 (scalar aggregation since in_dim==1) ----------------
__global__ void k_l1_edges(const int* __restrict__ src, const int* __restrict__ dst,
                           const float* __restrict__ dinv, const float* __restrict__ x,
                           float* s, int e) {
    int i = blockIdx.x * blockDim.x + threadIdx.x;
    if (i < e) {
        int u = src[i], v = dst[i];
        atomicAdd(&s[v], dinv[u] * dinv[v] * x[u]);
    }
}

__global__ void k_l1_feat(const float* __restrict__ s, const float* __restrict__ W1,
                          const float* __restrict__ b1, float* __restrict__ h1, int n) {
    int t = blockIdx.x * blockDim.x + threadIdx.x;
    if (t < n * HID) {
        int i = t >> 6, h = t & 63;
        float v = s[i] * W1[h] + b1[h];
        h1[t] = v > 0.0f ? v : 0.0f;
    }
}

// ---------------- layer 2 dense GEMM: t = h1[N,64] @ W2[64,64] via fp32 WMMA ----------------
// B stored in LDS K-pair interleaved: element (k, col) -> pair k>>1, slot col*2 + (k&1).
// Each B fragment is then a single aligned ds_load_b64 into an even VGPR pair.
__global__ __launch_bounds__(256)
void k_gemm_wmma(const float* __restrict__ h1, const float* __restrict__ W2,
                 float* __restrict__ t, int n_tiles) {
    __shared__ v2f sB[(HID / 2) * BPAIR_STRIDE];      // 32 K-pairs x 80 v2f (20 KB)
    {
        float* sf = (float*)sB;
        for (int idx = threadIdx.x; idx < HID * HID; idx += blockDim.x) {
            int k = idx >> 6, col = idx & 63;
            sf[(k >> 1) * (2 * BPAIR_STRIDE) + col * 2 + (k & 1)] = W2[idx];
        }
    }
    __syncthreads();

    const int wave = threadIdx.x >> 5;
    const int lane = threadIdx.x & 31;
    const int tile = blockIdx.x * 8 + wave;
    if (tile >= n_tiles) return;              // wave-uniform: EXEC stays all-ones

    const int row0 = tile * 16;
    const int m    = lane & 15;               // M index (A) / N index (B,C,D)
    const int half = lane >> 4;               // K-half selector

    v8f acc0 = {}, acc1 = {}, acc2 = {}, acc3 = {};
    const float* arow = h1 + (size_t)(row0 + m) * HID;

    for (int kk = 0; kk < HID; kk += 4) {
        const int ka = kk + 2 * half;         // lanes 0-15 -> K=kk,kk+1; 16-31 -> kk+2,kk+3
        v2f a;
        a.x = arow[ka];
        a.y = arow[ka + 1];

        const v2f* bp = &sB[(ka >> 1) * BPAIR_STRIDE + m];   // (W2[ka][col], W2[ka+1][col])
        acc0 = __builtin_amdgcn_wmma_f32_16x16x4_f32(false, a, false, bp[ 0], (short)0, acc0, false, false);
        acc1 = __builtin_amdgcn_wmma_f32_16x16x4_f32(false, a, false, bp[16], (short)0, acc1, false, false);
        acc2 = __builtin_amdgcn_wmma_f32_16x16x4_f32(false, a, false, bp[32], (short)0, acc2, false, false);
        acc3 = __builtin_amdgcn_wmma_f32_16x16x4_f32(false, a, false, bp[48], (short)0, acc3, false, false);
    }

    // D layout: VGPR r, lanes 0-15 -> M=r ; lanes 16-31 -> M=r+8 ; N = lane&15
    float* orow = t + (size_t)(row0 + half * 8) * HID;
#pragma unroll
    for (int r = 0; r < 8; ++r) {
        orow[(size_t)r * HID +      m] = acc0[r];
        orow[(size_t)r * HID + 16 + m] = acc1[r];
        orow[(size_t)r * HID + 32 + m] = acc2[r];
        orow[(size_t)r * HID + 48 + m] = acc3[r];
    }
}

// ---------------- layer 2 aggregation ----------------
__global__ void k_l2_init(const float* __restrict__ t, const float* __restrict__ dinv,
                          float* __restrict__ out2, int n) {
    int tid = blockIdx.x * blockDim.x + threadIdx.x;
    if (tid < n * HID) {
        float d = dinv[tid >> 6];
        out2[tid] = d * d * t[tid];           // self-loop contribution
    }
}

__global__ void k_l2_edges(const int* __restrict__ src, const int* __restrict__ dst,
                           const float* __restrict__ dinv, const float* __restrict__ t,
                           float* out2, int e) {
    int tid = blockIdx.x * blockDim.x + threadIdx.x;
    int ei  = tid >> 5;                       // 32 lanes per edge, float2 per lane
    if (ei >= e) return;
    int lane = tid & 31;
    int u = src[ei], v = dst[ei];
    float w = dinv[u] * dinv[v];
    const float* tr = t + (size_t)u * HID + lane * 2;
    float* orow = out2 + (size_t)v * HID + lane * 2;
    atomicAdd(&orow[0], w * tr[0]);
    atomicAdd(&orow[1], w * tr[1]);
}

// ---------------- pooling ----------------
__global__ void k_pool_init(float* pool, float* cnt) {
    int i = blockIdx.x * blockDim.x + threadIdx.x;
    if (i < N_GRAPH * HID) pool[i] = 0.0f;
    if (i < N_GRAPH) cnt[i] = 0.0f;
}

__global__ void k_relu_pool(const float* __restrict__ out2, const float* __restrict__ b2,
                            const int* __restrict__ batch, float* pool, float* cnt, int n) {
    int tid = blockIdx.x * blockDim.x + threadIdx.x;
    if (tid < n * HID) {
        int i = tid >> 6, h = tid & 63;
        float v = out2[tid] + b2[h];
        v = v > 0.0f ? v : 0.0f;
        int g = batch[i];
        atomicAdd(&pool[g * HID + h], v);
        if (h == 0) atomicAdd(&cnt[g], 1.0f);
    }
}

__global__ void k_final(const float* __restrict__ pool, const float* __restrict__ cnt,
                        float* __restrict__ out) {
    int tid = blockIdx.x * blockDim.x + threadIdx.x;
    if (tid < N_GRAPH * HID) {
        float c = cnt[tid >> 6];
        out[tid] = pool[tid] / fmaxf(c, 1.0f);
    }
}

extern "C" void kernel_launch(void* const* d_in, const int* in_sizes, int n_in,
                              void* d_out, int out_size, void* d_ws, size_t ws_size,
                              hipStream_t stream) {
    const float* x     = (const float*)d_in[0];
    const int*   edge  = (const int*)  d_in[1];   // [2, E] flat: row0=src, row1=dst
    const int*   batch = (const int*)  d_in[2];
    const float* W1    = (const float*)d_in[3];
    const float* b1    = (const float*)d_in[4];
    const float* W2    = (const float*)d_in[5];
    const float* b2    = (const float*)d_in[6];
    const int* src = edge;
    const int* dst = edge + N_EDGES;
    float* out = (float*)d_out;

    float* ws   = (float*)d_ws;
    float* dinv = ws;                               // [N]  deg -> dinv in place
    float* s    = dinv + N_NODES;                   // [N]  layer-1 scalar agg
    float* bufA = s + N_NODES;                      // [N*H] h1, later out2 accumulator
    float* bufB = bufA + (size_t)N_NODES * HID;     // [N*H] t = h1 @ W2
    float* pool = bufB + (size_t)N_NODES * HID;     // [G*H]
    float* cnt  = pool + N_GRAPH * HID;             // [G]

    const int TB = 256;
    k_init_deg     <<<(N_NODES + TB - 1) / TB, TB, 0, stream>>>(dinv, N_NODES);
    k_deg_edges    <<<(N_EDGES + TB - 1) / TB, TB, 0, stream>>>(dst, dinv, N_EDGES);
    k_dinv_selfloop<<<(N_NODES + TB - 1) / TB, TB, 0, stream>>>(dinv, x, s, N_NODES);
    k_l1_edges     <<<(N_EDGES + TB - 1) / TB, TB, 0, stream>>>(src, dst, dinv, x, s, N_EDGES);
    k_l1_feat      <<<(N_NODES * HID + TB - 1) / TB, TB, 0, stream>>>(s, W1, b1, bufA, N_NODES);

    const int n_tiles = N_NODES / 16;               // 12500, N divisible by 16
    k_gemm_wmma    <<<(n_tiles + 7) / 8, 256, 0, stream>>>(bufA, W2, bufB, n_tiles);

    k_l2_init      <<<(N_NODES * HID + TB - 1) / TB, TB, 0, stream>>>(bufB, dinv, bufA, N_NODES);
    {
        long long thr = (long long)N_EDGES * 32;
        k_l2_edges <<<(unsigned)((thr + TB - 1) / TB), TB, 0, stream>>>(src, dst, dinv, bufB, bufA, N_EDGES);
    }
    k_pool_init    <<<(N_GRAPH * HID + TB - 1) / TB, TB, 0, stream>>>(pool, cnt);
    k_relu_pool    <<<(N_NODES * HID + TB - 1) / TB, TB, 0, stream>>>(bufA, b2, batch, pool, cnt, N_NODES);
    k_final        <<<(N_GRAPH * HID + TB - 1) / TB, TB, 0, stream>>>(pool, cnt, out);
}